// DETRLoss_17051020165264
// MI455X (gfx1250) — compile-verified
//
#include <hip/hip_runtime.h>
#include <hip/hip_bf16.h>
#include <math.h>

#define NQ        100
#define BZ        64
#define NCLS      1203
#define NO_OBJ    1202
#define NROWS     (BZ * NQ)          // 6400
#define COSTN     (NQ * NQ)          // 10000 per batch
#define JV_INF    1e18

typedef __attribute__((ext_vector_type(2))) float        v2f;
typedef __attribute__((ext_vector_type(8))) float        v8f;
typedef __attribute__((ext_vector_type(4))) unsigned int v4u;
typedef __attribute__((ext_vector_type(4))) int          v4i;
typedef __attribute__((ext_vector_type(8))) int          v8i;

// ---------------------------------------------------------------------------
// Wave(32)-wide float sum using the CDNA5 matrix pipe:
// D = A x B + 0 with B = ones(4x16).  A layout (32-bit A, 16x4): lane L<16
// supplies A[L][0..1] from VGPR0/1, lane L>=16 supplies A[L-16][2..3].
// With a = {p, 0}: D[m][n] = p[m] + p[m+16] for every n.
// C/D layout: lane n holds D[v][n] (v=0..7) / D[v+8][n-16].  Summing the 8
// accumulator elements per lane gives each half-wave half the total; one
// shfl_xor(16) completes the 32-lane reduction.
// Requires EXEC == all ones (call only from convergent full-wave code).
// ---------------------------------------------------------------------------
__device__ __forceinline__ float wave_sum_f32_wmma(float p) {
    v2f a; a.x = p;    a.y = 0.0f;
    v2f b; b.x = 1.0f; b.y = 1.0f;
    v8f c = {};
    c = __builtin_amdgcn_wmma_f32_16x16x4_f32(false, a, false, b,
                                              (short)0, c, false, false);
    float s = c[0] + c[1] + c[2] + c[3] + c[4] + c[5] + c[6] + c[7];
    s += __shfl_xor(s, 16, 32);
    return s;
}

__device__ __forceinline__ double wave_sum_f64(double x) {
    for (int off = 16; off > 0; off >>= 1) x += __shfl_xor(x, off, 32);
    return x;
}

__device__ __forceinline__ int wave_sum_i32(int x) {
    for (int off = 16; off > 0; off >>= 1) x += __shfl_xor(x, off, 32);
    return x;
}

// ---------------------------------------------------------------------------
// Tensor Data Mover: DMA a contiguous run of 400 dwords (one batch's boxes,
// 1600 B) from global memory into LDS.  D# built per ISA 8.3/8.4:
//   group0: count=1 | lds_addr | global_addr[56:0] | type=2
//   group1: data_size=4B, tensor_dim0=400, tensor_dim1=1, tile_dim0=400,
//           tile_dim1=1, tensor_dim0_stride=400, tensor_dim1_stride=400
//   groups 2/3: zero (dims 2..4 unused)
// lds_off must be the LDS byte address = low 32 bits of the generic pointer
// (shared aperture occupies only the high dword).  Issue from ONE wave per
// block (EXEC is ignored by TDM; avoids duplicate DMAs), then
// s_wait_tensorcnt 0 + block barrier before consumers read LDS.
// ---------------------------------------------------------------------------
#if __has_builtin(__builtin_amdgcn_tensor_load_to_lds)
#define HAVE_TDM 1
__device__ __forceinline__ void tdm_load_400dw(const float* gsrc, unsigned lds_off) {
    unsigned long long ga = (unsigned long long)(size_t)gsrc;
    v4u g0 = { 1u,                                  // count=1, is_restore=0
               lds_off,                             // lds_addr
               (unsigned)(ga & 0xFFFFFFFFu),        // global_addr[31:0]
               (unsigned)((ga >> 32) & 0x01FFFFFFu) // global_addr[56:32]
                 | 0x80000000u };                   // type=2 ("image")
    v8i g1 = { (int)(2u << 16),     // workgroup_mask=0 | data_size=2 (4B)
               (int)(400u << 16),   // atomic_barrier_addr=0 | tensor_dim0.lo=400
               (int)(1u << 16),     // tensor_dim0.hi=0      | tensor_dim1.lo=1
               (int)(400u << 16),   // tensor_dim1.hi=0      | tile_dim0=400
               1,                   // tile_dim1=1 | tile_dim2=0
               400,                 // tensor_dim0_stride[31:0]=400
               (int)(400u << 16),   // stride0[47:32]=0 | tensor_dim1_stride.lo=400
               0 };                 // tensor_dim1_stride[47:16]=0
    v4i gz = { 0, 0, 0, 0 };
#if defined(__clang_major__) && (__clang_major__ >= 23)
    v8i gz8 = { 0, 0, 0, 0, 0, 0, 0, 0 };
    __builtin_amdgcn_tensor_load_to_lds(g0, g1, gz, gz, gz8, 0);
#else
    __builtin_amdgcn_tensor_load_to_lds(g0, g1, gz, gz, 0);
#endif
}
#else
#define HAVE_TDM 0
#endif

// ---------------------------------------------------------------------------
// Kernel A: per-row softmax statistics.
//   double max/sumexp  -> matcher (reference casts to float64 there)
//   float  max/sumexp  -> loss softmax (fp32 in jax)
//   argmax(logits) == argmax(softmax), first-index tie-break like jnp.argmax
// One wave per row, 8 waves per block, 800 blocks = 6400 rows exactly.
// ---------------------------------------------------------------------------
__global__ void softmax_stats_kernel(const float* __restrict__ lab_preds,
                                     double* __restrict__ rowMaxD,
                                     double* __restrict__ rowSumD,
                                     float*  __restrict__ rowMax32,
                                     float*  __restrict__ rowSum32,
                                     int*    __restrict__ rowArgmax) {
    const int wave = threadIdx.x >> 5;
    const int lane = threadIdx.x & 31;
    const int r    = blockIdx.x * 8 + wave;
    const float* __restrict__ base = lab_preds + (size_t)r * NCLS;

    // pass 1: max + argmax (strided-per-lane keeps first-occurrence semantics)
    float m = -3.4e38f;
    int   mi = 0x7fffffff;
    for (int c = lane; c < NCLS; c += 32) {
        float x = base[c];
        if (x > m) { m = x; mi = c; }
    }
    for (int off = 16; off > 0; off >>= 1) {
        float om = __shfl_xor(m, off, 32);
        int   oi = __shfl_xor(mi, off, 32);
        if (om > m || (om == m && oi < mi)) { m = om; mi = oi; }
    }
    const double md = (double)m;

    // pass 2: sum of exp in fp32 (loss) and fp64 (matcher)
    float  s32 = 0.0f;
    double sd  = 0.0;
    for (int c = lane; c < NCLS; c += 32) {
        float x = base[c];
        s32 += expf(x - m);
        sd  += exp((double)x - md);
    }
    // convergent here: full wave, loops reconverged -> WMMA legal (EXEC all 1s)
    float  s32t = wave_sum_f32_wmma(s32);
    double sdt  = wave_sum_f64(sd);

    if (lane == 0) {
        rowMaxD[r]   = md;
        rowSumD[r]   = sdt;
        rowMax32[r]  = m;
        rowSum32[r]  = s32t;
        rowArgmax[r] = mi;
    }
}

// ---------------------------------------------------------------------------
// Kernel B: per-batch 100x100 cost matrix (double, like reference):
//   C[i][j] = -softmax64(lab_preds[b,i])[labs[b,j]]
//           + L1(bbox_preds[b,i], bbox[b,j])
//           + (1 - IoU(xyxy(pred_i), xyxy(gt_j)))
// One block (128 thr) per batch; boxes staged in LDS via the TDM.
// ---------------------------------------------------------------------------
__global__ void cost_kernel(const int*   __restrict__ labs,
                            const float* __restrict__ lab_preds,
                            const float* __restrict__ bbox,
                            const float* __restrict__ bbox_preds,
                            const double* __restrict__ rowMaxD,
                            const double* __restrict__ rowSumD,
                            double* __restrict__ cost) {
    const int b   = blockIdx.x;
    const int tid = threadIdx.x;
    __shared__ int    s_lab[NQ];
    __shared__ float  s_gt[NQ * 4];
    __shared__ float  s_pr[NQ * 4];
    __shared__ double s_max[NQ];
    __shared__ double s_sum[NQ];

    for (int t = tid; t < NQ; t += 128) {
        s_lab[t] = labs[b * NQ + t];
        s_max[t] = rowMaxD[b * NQ + t];
        s_sum[t] = rowSumD[b * NQ + t];
    }
#if HAVE_TDM
    if (tid < 32) {   // wave-uniform branch: wave 0 issues the two tile DMAs
        tdm_load_400dw(bbox       + (size_t)b * NQ * 4,
                       (unsigned)(size_t)&s_gt[0]);
        tdm_load_400dw(bbox_preds + (size_t)b * NQ * 4,
                       (unsigned)(size_t)&s_pr[0]);
        __builtin_amdgcn_s_wait_tensorcnt(0);
    }
#else
    for (int t = tid; t < NQ * 4; t += 128) {
        s_gt[t] = bbox[b * NQ * 4 + t];
        s_pr[t] = bbox_preds[b * NQ * 4 + t];
    }
#endif
    __syncthreads();

    double* __restrict__ Cb = cost + (size_t)b * COSTN;
    for (int t = tid; t < COSTN; t += 128) {
        const int i = t / NQ;      // prediction row
        const int j = t - i * NQ;  // target column
        const int lab = s_lab[j];
        const float* gaddr = lab_preds + (size_t)(b * NQ + i) * NCLS + lab;
        __builtin_prefetch(gaddr, 0, 1);          // global_prefetch_b8
        const double pd = exp((double)(*gaddr) - s_max[i]) / s_sum[i];

        double l1 = 0.0;
        #pragma unroll
        for (int k = 0; k < 4; ++k)
            l1 += fabs((double)s_pr[i * 4 + k] - (double)s_gt[j * 4 + k]);

        const double pcx = s_pr[i*4+0], pcy = s_pr[i*4+1];
        const double pw  = s_pr[i*4+2], ph  = s_pr[i*4+3];
        const double gcx = s_gt[j*4+0], gcy = s_gt[j*4+1];
        const double gw  = s_gt[j*4+2], gh  = s_gt[j*4+3];
        const double ax0 = pcx - pw*0.5, ay0 = pcy - ph*0.5;
        const double ax1 = pcx + pw*0.5, ay1 = pcy + ph*0.5;
        const double bx0 = gcx - gw*0.5, by0 = gcy - gh*0.5;
        const double bx1 = gcx + gw*0.5, by1 = gcy + gh*0.5;
        const double areaA = (ax1-ax0)*(ay1-ay0);
        const double areaB = (bx1-bx0)*(by1-by0);
        const double w  = fmax(fmin(ax1,bx1) - fmax(ax0,bx0), 0.0);
        const double h  = fmax(fmin(ay1,by1) - fmax(ay0,by0), 0.0);
        const double inter = w * h;
        const double iou   = inter / fmax(areaA + areaB - inter, 1e-9);

        Cb[t] = -pd + l1 + (1.0 - iou);
    }
}

// ---------------------------------------------------------------------------
// Kernel C: exact Jonker-Volgenant Hungarian (mirrors the reference _lsa,
// including np.argmin first-index tie-breaking).  One block per batch; the
// O(n) column-relax and argmin steps are block-parallel, bookkeeping serial.
// ---------------------------------------------------------------------------
__global__ void hungarian_kernel(const double* __restrict__ cost,
                                 int* __restrict__ gtIdx) {
    const int b   = blockIdx.x;
    const int tid = threadIdx.x;
    const double* __restrict__ C = cost + (size_t)b * COSTN;

    __shared__ double u[NQ + 1], v[NQ + 1], minv[NQ + 1];
    __shared__ int    p[NQ + 1], way[NQ + 1], used[NQ + 1];
    __shared__ double redv[128];
    __shared__ int    redi[128];
    __shared__ double s_delta;
    __shared__ int    s_j0, s_j1, s_i0, s_done;

    for (int j = tid; j <= NQ; j += 128) { u[j] = 0.0; v[j] = 0.0; p[j] = 0; way[j] = 0; }
    __syncthreads();

    for (int i = 1; i <= NQ; ++i) {
        for (int j = tid; j <= NQ; j += 128) { minv[j] = JV_INF; used[j] = 0; }
        if (tid == 0) { p[0] = i; s_j0 = 0; }
        __syncthreads();

        while (true) {
            if (tid == 0) { used[s_j0] = 1; s_i0 = p[s_j0]; }
            __syncthreads();
            const int i0 = s_i0;
            const int j0 = s_j0;

            // relax unused columns; gather candidates for argmin
            double lv = JV_INF;
            int    li = NQ + 1;
            if (tid < NQ) {
                const int j = tid + 1;
                if (!used[j]) {
                    double cur = C[(size_t)(i0 - 1) * NQ + (j - 1)] - u[i0] - v[j];
                    if (cur < minv[j]) { minv[j] = cur; way[j] = j0; }
                    lv = minv[j];
                    li = j;
                }
            }
            redv[tid] = lv; redi[tid] = li;
            __syncthreads();
            for (int off = 64; off > 0; off >>= 1) {
                if (tid < off) {
                    double ov = redv[tid + off];
                    int    oi = redi[tid + off];
                    if (ov < redv[tid] || (ov == redv[tid] && oi < redi[tid])) {
                        redv[tid] = ov; redi[tid] = oi;
                    }
                }
                __syncthreads();
            }
            if (tid == 0) { s_j1 = redi[0]; s_delta = redv[0]; }
            __syncthreads();
            const double delta = s_delta;
            const int    j1    = s_j1;

            // dual updates (p[j] distinct over used columns -> race-free)
            if (tid <= NQ) {
                if (used[tid])      { u[p[tid]] += delta; v[tid] -= delta; }
                else if (tid >= 1)  { minv[tid] -= delta; }
            }
            __syncthreads();
            if (tid == 0) { s_j0 = j1; s_done = (p[j1] == 0); }
            __syncthreads();
            if (s_done) break;
        }

        if (tid == 0) {            // augment along alternating path
            int j0 = s_j0;
            while (j0 != 0) { int j1 = way[j0]; p[j0] = p[j1]; j0 = j1; }
        }
        __syncthreads();
    }

    if (tid >= 1 && tid <= NQ)
        gtIdx[b * NQ + (p[tid] - 1)] = tid - 1;
}

// ---------------------------------------------------------------------------
// Kernel D: final scalar loss (fp32, like the jax side).
//   loss = mean(-p32[new_lab]) + sum_masked(L1) + sum_masked(1-IoU)/max(cnt,1)
// Single block; per-wave reductions on the matrix pipe, 8 wave partials
// combined by thread 0.
// ---------------------------------------------------------------------------
__global__ void loss_kernel(const int*   __restrict__ labs,
                            const float* __restrict__ lab_preds,
                            const float* __restrict__ bbox,
                            const float* __restrict__ bbox_preds,
                            const float* __restrict__ rowMax32,
                            const float* __restrict__ rowSum32,
                            const int*   __restrict__ rowArgmax,
                            const int*   __restrict__ gtIdx,
                            float* __restrict__ out) {
    const int tid  = threadIdx.x;
    const int wv   = tid >> 5;
    const int lane = tid & 31;
    float  accP = 0.0f, accL1 = 0.0f, accIou = 0.0f;
    int    accCnt = 0;

    for (int idx = tid; idx < NROWS; idx += 256) {   // 6400/256 = 25 exact
        const int b = idx / NQ;
        const int g = gtIdx[idx];
        const int nl = labs[b * NQ + g];

        const float pnew = expf(lab_preds[(size_t)idx * NCLS + nl] - rowMax32[idx])
                           / rowSum32[idx];
        accP += pnew;

        const bool msk = (nl != NO_OBJ) && (nl == rowArgmax[idx]);
        if (msk) {
            ++accCnt;
            const float* nb = bbox       + (size_t)(b * NQ + g) * 4;
            const float* pb = bbox_preds + (size_t)idx * 4;
            accL1 += fabsf(nb[0]-pb[0]) + fabsf(nb[1]-pb[1])
                   + fabsf(nb[2]-pb[2]) + fabsf(nb[3]-pb[3]);

            const float ax0 = nb[0]-nb[2]*0.5f, ay0 = nb[1]-nb[3]*0.5f;
            const float ax1 = nb[0]+nb[2]*0.5f, ay1 = nb[1]+nb[3]*0.5f;
            const float bx0 = pb[0]-pb[2]*0.5f, by0 = pb[1]-pb[3]*0.5f;
            const float bx1 = pb[0]+pb[2]*0.5f, by1 = pb[1]+pb[3]*0.5f;
            const float w = fmaxf(fminf(ax1,bx1) - fmaxf(ax0,bx0), 0.0f);
            const float h = fmaxf(fminf(ay1,by1) - fmaxf(ay0,by0), 0.0f);
            const float inter = w * h;
            const float uni = (ax1-ax0)*(ay1-ay0) + (bx1-bx0)*(by1-by0) - inter;
            accIou += 1.0f - inter / fmaxf(uni, 1e-9f);
        }
    }

    // convergent here (fixed trip count, mask branch reconverged): WMMA legal
    const float wp = wave_sum_f32_wmma(accP);
    const float wl = wave_sum_f32_wmma(accL1);
    const float wi = wave_sum_f32_wmma(accIou);
    const int   wc = wave_sum_i32(accCnt);

    __shared__ float sp[8], sl[8], si[8];
    __shared__ int   sc[8];
    if (lane == 0) { sp[wv] = wp; sl[wv] = wl; si[wv] = wi; sc[wv] = wc; }
    __syncthreads();
    if (tid == 0) {
        float tp = 0.f, tl = 0.f, ti = 0.f; int tc = 0;
        #pragma unroll
        for (int k = 0; k < 8; ++k) { tp += sp[k]; tl += sl[k]; ti += si[k]; tc += sc[k]; }
        const float lossLabel = -tp / (float)NROWS;
        const int   cnt = tc > 1 ? tc : 1;
        out[0] = lossLabel + tl + ti / (float)cnt;
    }
}

// ---------------------------------------------------------------------------
extern "C" void kernel_launch(void* const* d_in, const int* in_sizes, int n_in,
                              void* d_out, int out_size, void* d_ws, size_t ws_size,
                              hipStream_t stream) {
    const int*   labs       = (const int*)  d_in[0];
    const float* lab_preds  = (const float*)d_in[1];
    const float* bbox       = (const float*)d_in[2];
    const float* bbox_preds = (const float*)d_in[3];

    char* ws = (char*)d_ws;
    double* rowMaxD  = (double*)ws;  ws += (size_t)NROWS * sizeof(double);
    double* rowSumD  = (double*)ws;  ws += (size_t)NROWS * sizeof(double);
    double* cost     = (double*)ws;  ws += (size_t)BZ * COSTN * sizeof(double);
    float*  rowMax32 = (float*)ws;   ws += (size_t)NROWS * sizeof(float);
    float*  rowSum32 = (float*)ws;   ws += (size_t)NROWS * sizeof(float);
    int*    rowArgmax= (int*)ws;     ws += (size_t)NROWS * sizeof(int);
    int*    gtIdx    = (int*)ws;     ws += (size_t)NROWS * sizeof(int);

    softmax_stats_kernel<<<NROWS / 8, 256, 0, stream>>>(
        lab_preds, rowMaxD, rowSumD, rowMax32, rowSum32, rowArgmax);

    cost_kernel<<<BZ, 128, 0, stream>>>(
        labs, lab_preds, bbox, bbox_preds, rowMaxD, rowSumD, cost);

    hungarian_kernel<<<BZ, 128, 0, stream>>>(cost, gtIdx);

    loss_kernel<<<1, 256, 0, stream>>>(
        labs, lab_preds, bbox, bbox_preds, rowMax32, rowSum32, rowArgmax,
        gtIdx, (float*)d_out);
}